// GyroidicConstraintProjector_86251533238602
// MI455X (gfx1250) — compile-verified
//
#include <hip/hip_runtime.h>
#include <stdint.h>

// ---------------------------------------------------------------------------
// Gyroid constraint projector for MI455X (gfx1250).
// Transcendental-bound elementwise iteration; async global<->LDS staging.
// Clamps via v_med3_f32 (no NaN-canonicalize overhead; inputs provably finite).
// ---------------------------------------------------------------------------

#define TPB            256
#define TRI_PER_THR    4
#define FLOATS_PER_BLK (TPB * TRI_PER_THR * 3)    // 3072 floats = 12 KB / block
#define VEC_PER_THR    (FLOATS_PER_BLK / 4 / TPB) // 3 x b128 per thread

#define INV_2PI 0.15915494309189535f

__device__ __forceinline__ float hsin(float x) { return __builtin_amdgcn_sinf(x * INV_2PI); }
__device__ __forceinline__ float hcos(float x) { return __builtin_amdgcn_cosf(x * INV_2PI); }
// Single v_med3_f32 clamp; valid because no NaNs can occur in this pipeline.
__device__ __forceinline__ float clampf(float v, float lo, float hi) {
  return __builtin_amdgcn_fmed3f(v, lo, hi);
}

// One gradient-descent step of the gyroid projection.
__device__ __forceinline__ void gyroid_step(float& x, float& y, float& z) {
  float sx = hsin(x), cx = hcos(x);
  float sy = hsin(y), cy = hcos(y);
  float sz = hsin(z), cz = hcos(z);
  // cv = sin(x)cos(y) + sin(y)cos(z) + sin(z)cos(x)
  float cv  = __builtin_fmaf(sx, cy, __builtin_fmaf(sy, cz, sz * cx));
  float gx  = __builtin_fmaf(-sz, sx, cx * cy);
  float gy  = __builtin_fmaf(-sx, sy, cy * cz);
  float gz  = __builtin_fmaf(-sy, sz, cz * cx);
  float gn2 = __builtin_fmaf(gx, gx, __builtin_fmaf(gy, gy, gz * gz));
  float inv = __builtin_amdgcn_rsqf(gn2 + 1e-16f);    // ~ 1/(||g||+eps)
  float st  = clampf(0.1f * cv, -0.5f, 0.5f) * inv;   // step / ||g||
  x = clampf(__builtin_fmaf(-st, gx, x), -5.0f, 5.0f);
  y = clampf(__builtin_fmaf(-st, gy, y), -5.0f, 5.0f);
  z = clampf(__builtin_fmaf(-st, gz, z), -5.0f, 5.0f);
}

__global__ __launch_bounds__(TPB)
void gyroid_tile_kernel(const float* __restrict__ in, float* __restrict__ out) {
  __shared__ float tile[FLOATS_PER_BLK];
  const int      t       = threadIdx.x;
  const size_t   blkBase = (size_t)blockIdx.x * FLOATS_PER_BLK;
  const uint64_t gsrc    = (uint64_t)(uintptr_t)(in + blkBase);
  const uint64_t gdst    = (uint64_t)(uintptr_t)(out + blkBase);
  const unsigned ldsBase = (unsigned)(uintptr_t)(&tile[0]); // low 32 bits = LDS byte addr

  // ---- Stage in: async global -> LDS, coalesced b128 per lane --------------
#pragma unroll
  for (int k = 0; k < VEC_PER_THR; ++k) {
    unsigned voff = (unsigned)((t + k * TPB) * 16);
    unsigned la   = ldsBase + voff;
    asm volatile("global_load_async_to_lds_b128 %0, %1, %2"
                 :: "v"(la), "v"(voff), "s"(gsrc) : "memory");
  }
  asm volatile("s_wait_asynccnt 0" ::: "memory");
  __syncthreads();

  // ---- Load my 4 triples (12 contiguous floats per thread, lane stride 12) ----------
  float X[TRI_PER_THR], Y[TRI_PER_THR], Z[TRI_PER_THR];
  const int base = t * (TRI_PER_THR * 3);
#pragma unroll
  for (int i = 0; i < TRI_PER_THR; ++i) {
    X[i] = clampf(tile[base + 3 * i + 0], -3.0f, 3.0f);
    Y[i] = clampf(tile[base + 3 * i + 1], -3.0f, 3.0f);
    Z[i] = clampf(tile[base + 3 * i + 2], -3.0f, 3.0f);
  }

  // ---- 50 fixed iterations; 4 independent triples give trans/VALU co-exec --
#pragma unroll 1
  for (int it = 0; it < 50; ++it) {
#pragma unroll
    for (int i = 0; i < TRI_PER_THR; ++i) {
      gyroid_step(X[i], Y[i], Z[i]);
    }
  }

  // ---- Write back to LDS, then async LDS -> global -------------------------
#pragma unroll
  for (int i = 0; i < TRI_PER_THR; ++i) {
    tile[base + 3 * i + 0] = X[i];
    tile[base + 3 * i + 1] = Y[i];
    tile[base + 3 * i + 2] = Z[i];
  }
  __syncthreads();

#pragma unroll
  for (int k = 0; k < VEC_PER_THR; ++k) {
    unsigned voff = (unsigned)((t + k * TPB) * 16);
    unsigned la   = ldsBase + voff;
    asm volatile("global_store_async_from_lds_b128 %0, %1, %2"
                 :: "v"(voff), "v"(la), "s"(gdst) : "memory");
  }
  asm volatile("s_wait_asynccnt 0" ::: "memory");
}

// Generic tail path (never fires for the 4096x6144 shape; kept for safety).
__global__ __launch_bounds__(TPB)
void gyroid_tail_kernel(const float* __restrict__ in, float* __restrict__ out,
                        long long startFloat, long long nTriples) {
  long long j = (long long)blockIdx.x * TPB + threadIdx.x;
  if (j >= nTriples) return;
  long long o = startFloat + 3 * j;
  float x = clampf(in[o + 0], -3.0f, 3.0f);
  float y = clampf(in[o + 1], -3.0f, 3.0f);
  float z = clampf(in[o + 2], -3.0f, 3.0f);
#pragma unroll 1
  for (int it = 0; it < 50; ++it) gyroid_step(x, y, z);
  out[o + 0] = x;
  out[o + 1] = y;
  out[o + 2] = z;
}

extern "C" void kernel_launch(void* const* d_in, const int* in_sizes, int n_in,
                              void* d_out, int out_size, void* d_ws, size_t ws_size,
                              hipStream_t stream) {
  const float* in  = (const float*)d_in[0];
  float*       out = (float*)d_out;
  const long long total = (long long)in_sizes[0];           // flat float count

  const long long fullBlocks = total / FLOATS_PER_BLK;      // 8192 for 4096x6144
  if (fullBlocks > 0) {
    gyroid_tile_kernel<<<dim3((unsigned)fullBlocks), dim3(TPB), 0, stream>>>(in, out);
  }
  const long long remFloats  = total - fullBlocks * FLOATS_PER_BLK;
  const long long remTriples = remFloats / 3;
  if (remTriples > 0) {
    const unsigned tailGrid = (unsigned)((remTriples + TPB - 1) / TPB);
    gyroid_tail_kernel<<<dim3(tailGrid), dim3(TPB), 0, stream>>>(
        in, out, fullBlocks * (long long)FLOATS_PER_BLK, remTriples);
  }
}